// MLA_Inference_24008867184767
// MI455X (gfx1250) — compile-verified
//
#include <hip/hip_runtime.h>
#include <hip/hip_bf16.h>
#include <math.h>

#define B_    32
#define T_    4096
#define PAST_ 4095
#define DM    2048
#define H_    16
#define QL    1536
#define C_    512
#define R_    64
#define KVM   576      // C_ + R_
#define HR    1024     // H_ * R_
#define SCALE 0.07216878364870322f   // 1/sqrt(192)
#define LNIF  0.28782313663f         // ln(10000)/32

typedef float v2f __attribute__((ext_vector_type(2)));
typedef float v8f __attribute__((ext_vector_type(8)));
typedef unsigned int u32x4 __attribute__((ext_vector_type(4)));
typedef int i32x4 __attribute__((ext_vector_type(4)));
typedef int i32x8 __attribute__((ext_vector_type(8)));

__device__ __forceinline__ v8f wmma4(v2f a, v2f b, v8f c) {
  // D = A(16x4 f32) * B(4x16 f32) + C(16x16 f32)
  return __builtin_amdgcn_wmma_f32_16x16x4_f32(false, a, false, b, (short)0, c,
                                               false, false);
}

__device__ __forceinline__ float waveMax(float x) {
  for (int o = 16; o; o >>= 1) x = fmaxf(x, __shfl_xor(x, o, 32));
  return x;
}
__device__ __forceinline__ float waveSum(float x) {
  for (int o = 16; o; o >>= 1) x += __shfl_xor(x, o, 32);
  return x;
}

// ---------------------------------------------------------------------------
// TDM: DMA one [32 x 512] f32 tile (row-major, stride 512) from global -> LDS.
// D# group0: count=1, lds_addr, 57-bit global addr, type=2 ("image").
// D# group1: data_size=4B, tensor_dim0=512, tensor_dim1=rows_avail (OOB rows
//            read back as zero), tile_dim0=512, tile_dim1=32, stride0=512.
// Groups 2/3 zero => 2-D descriptor.  Tracked by TENSORcnt.
// ---------------------------------------------------------------------------
__device__ __forceinline__ void tdm_load_kv_tile(unsigned lds_off_bytes,
                                                 const float* gsrc,
                                                 unsigned rows_avail) {
  unsigned long long ga = (unsigned long long)(uintptr_t)gsrc;
  u32x4 g0 = { 1u,                                   // count=1, user desc
               lds_off_bytes,                        // lds_addr
               (unsigned)(ga & 0xFFFFFFFFull),       // global_addr[31:0]
               (unsigned)((ga >> 32) & 0x1FFFFFFull) | (2u << 30) };  // +type=2
  i32x8 g1 = { (int)(2u << 16),                      // data_size = 4 bytes
               (int)(512u << 16),                    // tensor_dim0[15:0]
               (int)(rows_avail << 16),              // tensor_dim0[31:16]=0 | tensor_dim1[15:0]
               (int)(512u << 16),                    // tensor_dim1[31:16]=0 | tile_dim0=512
               32,                                   // tile_dim1=32, tile_dim2=0
               512,                                  // tensor_dim0_stride[31:0]
               0, 0 };
  i32x4 z4 = { 0, 0, 0, 0 };
#if __clang_major__ >= 23
  i32x8 z8 = { 0, 0, 0, 0, 0, 0, 0, 0 };
  __builtin_amdgcn_tensor_load_to_lds(g0, g1, z4, z4, z8, 0);
#else
  __builtin_amdgcn_tensor_load_to_lds(g0, g1, z4, z4, 0);
#endif
}

// ---------------------------------------------------------------------------
// Kernel 1: q_pre[32,1536] = hidden @ w_q_a^T ; kv_mixed[32,576] = hidden @ w_kv_a^T
// One fused GEMM: M=32, N=2112, K=2048.  132 N-tiles, 1 wave per N-tile.
// ---------------------------------------------------------------------------
__global__ void __launch_bounds__(128) k_proj1(
    const float* __restrict__ hidden, const float* __restrict__ w_q_a,
    const float* __restrict__ w_kv_a, float* __restrict__ q_pre,
    float* __restrict__ kv_mixed) {
  int wave = blockIdx.x * (blockDim.x >> 5) + (threadIdx.x >> 5);  // 0..131
  int lane = threadIdx.x & 31, g = lane >> 4, m = lane & 15;
  int n0 = wave * 16;
  int ncol = n0 + m;
  const float* wrow = (ncol < QL) ? w_q_a + (size_t)ncol * DM
                                  : w_kv_a + (size_t)(ncol - QL) * DM;
  const float* a0p = hidden + (size_t)m * DM;         // batch rows 0..15
  const float* a1p = hidden + (size_t)(m + 16) * DM;  // batch rows 16..31
  v8f c0 = {}; v8f c1 = {};
  for (int k0 = 0; k0 < DM; k0 += 4) {
    int kk = k0 + 2 * g;
    v2f a0 = *(const v2f*)(a0p + kk);
    v2f a1 = *(const v2f*)(a1p + kk);
    v2f bb = *(const v2f*)(wrow + kk);
    c0 = wmma4(a0, bb, c0);
    c1 = wmma4(a1, bb, c1);
  }
#pragma unroll
  for (int v = 0; v < 8; v++) {
    int r0 = v + 8 * g;
    int col = n0 + m;
    if (col < QL) {
      q_pre[(size_t)r0 * QL + col]        = c0[v];
      q_pre[(size_t)(r0 + 16) * QL + col] = c1[v];
    } else {
      kv_mixed[(size_t)r0 * KVM + (col - QL)]        = c0[v];
      kv_mixed[(size_t)(r0 + 16) * KVM + (col - QL)] = c1[v];
    }
  }
}

// ---------------------------------------------------------------------------
// Kernel 2: layernorms + rope on new k.  One block per batch row.
// ---------------------------------------------------------------------------
__global__ void __launch_bounds__(256) k_ln(
    const float* __restrict__ q_pre, const float* __restrict__ kv_mixed,
    const int* __restrict__ pos_ids,
    const float* __restrict__ gq, const float* __restrict__ bq,
    const float* __restrict__ gkv, const float* __restrict__ bkv,
    float* __restrict__ q_latent, float* __restrict__ kv_new) {
  int b = blockIdx.x;
  __shared__ float r1[8], r2[8];
  __shared__ float mean_s, inv_s;
  int tid = threadIdx.x, lane = tid & 31, w = tid >> 5;

  // ---- LN over q_pre[b, 0:1536]
  const float* x = q_pre + (size_t)b * QL;
  float s = 0.f, s2 = 0.f;
  for (int i = tid; i < QL; i += 256) { float v = x[i]; s += v; s2 += v * v; }
  s = waveSum(s); s2 = waveSum(s2);
  if (lane == 0) { r1[w] = s; r2[w] = s2; }
  __syncthreads();
  if (tid == 0) {
    float t1 = 0.f, t2 = 0.f;
    for (int i = 0; i < 8; i++) { t1 += r1[i]; t2 += r2[i]; }
    float mean = t1 / QL;
    mean_s = mean;
    inv_s = rsqrtf(t2 / QL - mean * mean + 1e-5f);
  }
  __syncthreads();
  float mean = mean_s, inv = inv_s;
  for (int i = tid; i < QL; i += 256)
    q_latent[(size_t)b * QL + i] = (x[i] - mean) * inv * gq[i] + bq[i];
  __syncthreads();

  // ---- LN over kv_mixed[b, 0:512] -> kv_new[b, 0:512]
  const float* y = kv_mixed + (size_t)b * KVM;
  s = 0.f; s2 = 0.f;
  for (int i = tid; i < C_; i += 256) { float v = y[i]; s += v; s2 += v * v; }
  s = waveSum(s); s2 = waveSum(s2);
  if (lane == 0) { r1[w] = s; r2[w] = s2; }
  __syncthreads();
  if (tid == 0) {
    float t1 = 0.f, t2 = 0.f;
    for (int i = 0; i < 8; i++) { t1 += r1[i]; t2 += r2[i]; }
    float mean2 = t1 / C_;
    mean_s = mean2;
    inv_s = rsqrtf(t2 / C_ - mean2 * mean2 + 1e-5f);
  }
  __syncthreads();
  mean = mean_s; inv = inv_s;
  for (int i = tid; i < C_; i += 256)
    kv_new[(size_t)b * KVM + i] = (y[i] - mean) * inv * gkv[i] + bkv[i];

  // ---- rope on kv_mixed[b, 512:576] -> kv_new[b, 512:576]
  if (tid < R_) {
    float p = (float)pos_ids[b];
    int i = tid, j = i & 31;
    float ang = p * __expf(-(float)j * LNIF);
    float xv = y[C_ + i];
    float partner = (i < 32) ? -y[C_ + i + 32] : y[C_ + i - 32];
    kv_new[(size_t)b * KVM + C_ + i] = xv * cosf(ang) + partner * sinf(ang);
  }
}

// ---------------------------------------------------------------------------
// Kernel 3: qr_pre[32,1024] = q_latent @ w_q_rope^T   (M=32,N=1024,K=1536)
// ---------------------------------------------------------------------------
__global__ void __launch_bounds__(128) k_qrope_mm(
    const float* __restrict__ q_latent, const float* __restrict__ w_q_rope,
    float* __restrict__ qr_pre) {
  int wave = blockIdx.x * (blockDim.x >> 5) + (threadIdx.x >> 5);  // 0..63
  int lane = threadIdx.x & 31, g = lane >> 4, m = lane & 15;
  int n0 = wave * 16;
  const float* wrow = w_q_rope + (size_t)(n0 + m) * QL;
  const float* a0p = q_latent + (size_t)m * QL;
  const float* a1p = q_latent + (size_t)(m + 16) * QL;
  v8f c0 = {}; v8f c1 = {};
  for (int k0 = 0; k0 < QL; k0 += 4) {
    int kk = k0 + 2 * g;
    v2f a0 = *(const v2f*)(a0p + kk);
    v2f a1 = *(const v2f*)(a1p + kk);
    v2f bb = *(const v2f*)(wrow + kk);
    c0 = wmma4(a0, bb, c0);
    c1 = wmma4(a1, bb, c1);
  }
#pragma unroll
  for (int v = 0; v < 8; v++) {
    int r0 = v + 8 * g;
    qr_pre[(size_t)r0 * HR + n0 + m]        = c0[v];
    qr_pre[(size_t)(r0 + 16) * HR + n0 + m] = c1[v];
  }
}

// ---------------------------------------------------------------------------
// Kernel 3b: rope on q_rope heads.
// ---------------------------------------------------------------------------
__global__ void __launch_bounds__(256) k_rope_q(
    const float* __restrict__ qr_pre, const int* __restrict__ pos_ids,
    float* __restrict__ q_rope) {
  int b = blockIdx.x;
  float p = (float)pos_ids[b];
  for (int idx = threadIdx.x; idx < HR; idx += blockDim.x) {
    int i = idx & 63, j = i & 31;
    float ang = p * __expf(-(float)j * LNIF);
    float xv = qr_pre[(size_t)b * HR + idx];
    float partner = (i < 32) ? -qr_pre[(size_t)b * HR + idx + 32]
                             :  qr_pre[(size_t)b * HR + idx - 32];
    q_rope[(size_t)b * HR + idx] = xv * cosf(ang) + partner * sinf(ang);
  }
}

// ---------------------------------------------------------------------------
// Kernel 4: q_abs[b,h,c] = sum_q q_latent[b,q] * fused_q_k[h,q,c]
// Per h: M=32, N=512, K=1536.  512 wave-tiles total.
// ---------------------------------------------------------------------------
__global__ void __launch_bounds__(128) k_qabs(
    const float* __restrict__ q_latent, const float* __restrict__ fused_q_k,
    float* __restrict__ q_abs) {
  int wave = blockIdx.x * (blockDim.x >> 5) + (threadIdx.x >> 5);  // 0..511
  int h = wave >> 5, ntile = wave & 31;
  int lane = threadIdx.x & 31, g = lane >> 4, m = lane & 15;
  int n0 = ntile * 16;
  const float* fq = fused_q_k + (size_t)h * QL * C_;
  const float* a0p = q_latent + (size_t)m * QL;
  const float* a1p = q_latent + (size_t)(m + 16) * QL;
  v8f c0 = {}; v8f c1 = {};
  for (int k0 = 0; k0 < QL; k0 += 4) {
    int kk = k0 + 2 * g;
    v2f a0 = *(const v2f*)(a0p + kk);
    v2f a1 = *(const v2f*)(a1p + kk);
    v2f bb = { fq[(size_t)kk * C_ + n0 + m], fq[(size_t)(kk + 1) * C_ + n0 + m] };
    c0 = wmma4(a0, bb, c0);
    c1 = wmma4(a1, bb, c1);
  }
#pragma unroll
  for (int v = 0; v < 8; v++) {
    int b0 = v + 8 * g;
    q_abs[((size_t)b0 * H_ + h) * C_ + n0 + m]        = c0[v];
    q_abs[((size_t)(b0 + 16) * H_ + h) * C_ + n0 + m] = c1[v];
  }
}

// ---------------------------------------------------------------------------
// Kernel 5: split-T flash decode.  grid (4 splits, 32 batch), 8 waves.
// Pass A: WMMA scores staged in LDS (direct global B-fragments are fully
//         clause-coalesced b64 loads).
// Pass B: per-head max/exp/sum.
// Pass C: P x KV with KV tiles DMA'd to LDS by the Tensor Data Mover,
//         double-buffered (TENSORcnt + barriers).  Token 4095 patched from
//         kv_new.  LDS: 64KB scores + 2x64KB KV = 193KB of the 320KB WGP LDS.
// ---------------------------------------------------------------------------
__global__ void __launch_bounds__(256) k_attn(
    const float* __restrict__ q_abs, const float* __restrict__ q_rope,
    const float* __restrict__ kv_nope, const float* __restrict__ k_rope,
    const float* __restrict__ kv_new,
    float* __restrict__ m_part, float* __restrict__ s_part,
    float* __restrict__ o_part) {
  extern __shared__ float smem[];
  float* sc   = smem;              // [16][1028]
  float* kvb0 = smem + 16448;      // [32][512]
  float* kvb1 = kvb0 + 16384;      // [32][512]
  const unsigned KVB0_OFF = 16448u * 4u;
  const unsigned KVB1_OFF = KVB0_OFF + 65536u;
  const int SP = 1028;
  int b = blockIdx.y, sp = blockIdx.x;
  int tid = threadIdx.x, lane = tid & 31, w = tid >> 5, g = lane >> 4, m = lane & 15;
  int t0 = sp * 1024;
  const float* kvbase = kv_nope + (size_t)b * PAST_ * C_;

  // Kick off the first two KV tile DMAs; they land while pass A/B compute.
  if (w == 0) {
    tdm_load_kv_tile(KVB0_OFF, kvbase + (size_t)t0 * C_, (unsigned)(PAST_ - t0));
    tdm_load_kv_tile(KVB1_OFF, kvbase + (size_t)(t0 + 32) * C_,
                     (unsigned)(PAST_ - (t0 + 32)));
  }

  const float* qa = q_abs + ((size_t)b * H_ + m) * C_;   // A row = head m
  const float* qr = q_rope + ((size_t)b * H_ + m) * R_;

  // ---- Pass A: scores for 64 token-tiles (8 per wave)
  for (int tt = w; tt < 64; tt += 8) {
    int tg = t0 + tt * 16 + m;  // this lane's token (B-matrix row)
    const float* kvrow = (tg < PAST_) ? kv_nope + ((size_t)b * PAST_ + tg) * C_
                                      : kv_new + (size_t)b * KVM;
    const float* krrow = (tg < PAST_) ? k_rope + ((size_t)b * PAST_ + tg) * R_
                                      : kv_new + (size_t)b * KVM + C_;
    if (tg + 128 < PAST_)
      __builtin_prefetch(kv_nope + ((size_t)b * PAST_ + tg + 128) * C_, 0, 0);
    v8f acc = {};
    for (int k0 = 0; k0 < C_; k0 += 4) {
      int kk = k0 + 2 * g;
      v2f a  = *(const v2f*)(qa + kk);
      v2f bb = *(const v2f*)(kvrow + kk);
      acc = wmma4(a, bb, acc);
    }
    for (int k0 = 0; k0 < R_; k0 += 4) {
      int kk = k0 + 2 * g;
      v2f a  = *(const v2f*)(qr + kk);
      v2f bb = *(const v2f*)(krrow + kk);
      acc = wmma4(a, bb, acc);
    }
    int tloc = tt * 16 + m;
#pragma unroll
    for (int v = 0; v < 8; v++)
      sc[(v + 8 * g) * SP + tloc] = acc[v] * SCALE;
  }
  __syncthreads();

  // ---- Pass B: per-head max / exp / sum.  Wave w owns heads 2w, 2w+1.
#pragma unroll
  for (int hh = 0; hh < 2; hh++) {
    int h = 2 * w + hh;
    float mx = -3.4e38f;
    for (int i = lane; i < 1024; i += 32) mx = fmaxf(mx, sc[h * SP + i]);
    mx = waveMax(mx);
    float sum = 0.f;
    for (int i = lane; i < 1024; i += 32) {
      float e = __expf(sc[h * SP + i] - mx);
      sc[h * SP + i] = e;
      sum += e;
    }
    sum = waveSum(sum);
    if (lane == 0) {
      m_part[((size_t)b * 4 + sp) * H_ + h] = mx;
      s_part[((size_t)b * 4 + sp) * H_ + h] = sum;
    }
  }

  // ---- Pass C: O_partial[16,512] = P[16,1024] x KV[1024,512]
  // 32 chunks of 32 tokens; KV chunks arrive via TDM double buffering.
  v8f o0 = {}, o1 = {}, o2 = {}, o3 = {};
  int nb = w * 64;
  for (int c = 0; c < 32; c++) {
    if (w == 0) {
      if (c < 31) __builtin_amdgcn_s_wait_tensorcnt(1);  // chunk c landed
      else        __builtin_amdgcn_s_wait_tensorcnt(0);
    }
    __syncthreads();
    float* kvb = (c & 1) ? kvb1 : kvb0;
    if (sp == 3 && c == 31) {
      // last row of last chunk is the appended token: patch from kv_new
      for (int i = tid; i < C_; i += 256)
        kvb[31 * 512 + i] = kv_new[(size_t)b * KVM + i];
      __syncthreads();
    }
    int cbase = c * 32;
    for (int lr0 = 0; lr0 < 32; lr0 += 4) {
      int lr = lr0 + 2 * g;
      v2f a = *(const v2f*)(&sc[m * SP + cbase + lr]);  // P[head=m][..]
      const float* r0p = kvb + lr * 512;
      const float* r1p = r0p + 512;
      v2f b0 = { r0p[nb + m],      r1p[nb + m] };
      v2f b1 = { r0p[nb + 16 + m], r1p[nb + 16 + m] };
      v2f b2 = { r0p[nb + 32 + m], r1p[nb + 32 + m] };
      v2f b3 = { r0p[nb + 48 + m], r1p[nb + 48 + m] };
      o0 = wmma4(a, b0, o0);
      o1 = wmma4(a, b1, o1);
      o2 = wmma4(a, b2, o2);
      o3 = wmma4(a, b3, o3);
    }
    __syncthreads();  // everyone done with kvb before it is overwritten
    if (w == 0 && c + 2 < 32) {
      unsigned tn = (unsigned)(t0 + (c + 2) * 32);
      tdm_load_kv_tile((c & 1) ? KVB1_OFF : KVB0_OFF,
                       kvbase + (size_t)tn * C_, (unsigned)(PAST_ - tn));
    }
  }
  float* op = o_part + ((size_t)b * 4 + sp) * H_ * C_;
#pragma unroll
  for (int v = 0; v < 8; v++) {
    int h = v + 8 * g;
    op[(size_t)h * C_ + nb + m]      = o0[v];
    op[(size_t)h * C_ + nb + 16 + m] = o1[v];
    op[(size_t)h * C_ + nb + 32 + m] = o2[v];
    op[(size_t)h * C_ + nb + 48 + m] = o3[v];
  }
}

// ---------------------------------------------------------------------------
// Kernel 6: merge 4 split-T partials (log-sum-exp combine).
// ---------------------------------------------------------------------------
__global__ void __launch_bounds__(128) k_reduce(
    const float* __restrict__ m_part, const float* __restrict__ s_part,
    const float* __restrict__ o_part, float* __restrict__ attn) {
  int b = blockIdx.x >> 4, h = blockIdx.x & 15;
  float mv[4], mx = -3.4e38f;
#pragma unroll
  for (int s = 0; s < 4; s++) {
    mv[s] = m_part[((size_t)b * 4 + s) * H_ + h];
    mx = fmaxf(mx, mv[s]);
  }
  float e[4], denom = 0.f;
#pragma unroll
  for (int s = 0; s < 4; s++) {
    e[s] = __expf(mv[s] - mx);
    denom += s_part[((size_t)b * 4 + s) * H_ + h] * e[s];
  }
  float rd = 1.f / denom;
  for (int c = threadIdx.x; c < C_; c += blockDim.x) {
    float acc = 0.f;
#pragma unroll
    for (int s = 0; s < 4; s++)
      acc += o_part[(((size_t)b * 4 + s) * H_ + h) * C_ + c] * e[s];
    attn[((size_t)b * H_ + h) * C_ + c] = acc * rd;
  }
}

// ---------------------------------------------------------------------------
// Kernel 7: out[32,2048] = attn_flat[32,8192] @ fused_o_v_flat[8192,2048]
// ---------------------------------------------------------------------------
__global__ void __launch_bounds__(128) k_out(
    const float* __restrict__ attn, const float* __restrict__ fused_o_v,
    float* __restrict__ out) {
  int wave = blockIdx.x * (blockDim.x >> 5) + (threadIdx.x >> 5);  // 0..127
  int lane = threadIdx.x & 31, g = lane >> 4, m = lane & 15;
  int n0 = wave * 16;
  const float* a0p = attn + (size_t)m * (H_ * C_);
  const float* a1p = attn + (size_t)(m + 16) * (H_ * C_);
  v8f c0 = {}; v8f c1 = {};
  for (int k0 = 0; k0 < H_ * C_; k0 += 4) {
    int kk = k0 + 2 * g;
    v2f a0 = *(const v2f*)(a0p + kk);
    v2f a1 = *(const v2f*)(a1p + kk);
    v2f bb = { fused_o_v[(size_t)kk * DM + n0 + m],
               fused_o_v[(size_t)(kk + 1) * DM + n0 + m] };
    c0 = wmma4(a0, bb, c0);
    c1 = wmma4(a1, bb, c1);
  }
#pragma unroll
  for (int v = 0; v < 8; v++) {
    int r0 = v + 8 * g;
    out[(size_t)r0 * DM + n0 + m]        = c0[v];
    out[(size_t)(r0 + 16) * DM + n0 + m] = c1[v];
  }
}

// ---------------------------------------------------------------------------
extern "C" void kernel_launch(void* const* d_in, const int* in_sizes, int n_in,
                              void* d_out, int out_size, void* d_ws, size_t ws_size,
                              hipStream_t stream) {
  (void)in_sizes; (void)n_in; (void)out_size; (void)ws_size;
  const float* hidden   = (const float*)d_in[0];
  const int*   pos      = (const int*)  d_in[1];
  const float* kv_nope  = (const float*)d_in[2];
  const float* k_rope   = (const float*)d_in[3];
  const float* w_q_a    = (const float*)d_in[4];
  const float* ln_q_g   = (const float*)d_in[5];
  const float* ln_q_b   = (const float*)d_in[6];
  const float* w_q_rope = (const float*)d_in[7];
  const float* w_kv_a   = (const float*)d_in[8];
  const float* ln_kv_g  = (const float*)d_in[9];
  const float* ln_kv_b  = (const float*)d_in[10];
  const float* fused_qk = (const float*)d_in[11];
  const float* fused_ov = (const float*)d_in[12];
  float* out = (float*)d_out;

  float* ws       = (float*)d_ws;
  float* q_pre    = ws;                  // 32*1536
  float* kv_mixed = q_pre + 49152;       // 32*576
  float* q_latent = kv_mixed + 18432;    // 32*1536
  float* kv_new   = q_latent + 49152;    // 32*576
  float* qr_pre   = kv_new + 18432;      // 32*1024
  float* q_rope_  = qr_pre + 32768;      // 32*1024
  float* q_abs    = q_rope_ + 32768;     // 32*16*512
  float* m_part   = q_abs + 262144;      // 32*4*16
  float* s_part   = m_part + 2048;       // 32*4*16
  float* o_part   = s_part + 2048;       // 32*4*16*512
  float* attn     = o_part + 1048576;    // 32*16*512

  k_proj1<<<33, 128, 0, stream>>>(hidden, w_q_a, w_kv_a, q_pre, kv_mixed);
  k_ln<<<32, 256, 0, stream>>>(q_pre, kv_mixed, pos, ln_q_g, ln_q_b,
                               ln_kv_g, ln_kv_b, q_latent, kv_new);
  k_qrope_mm<<<16, 128, 0, stream>>>(q_latent, w_q_rope, qr_pre);
  k_rope_q<<<32, 256, 0, stream>>>(qr_pre, pos, q_rope_);
  k_qabs<<<128, 128, 0, stream>>>(q_latent, fused_qk, q_abs);
  dim3 ag(4, 32);
  size_t lds_bytes = (16448u + 2u * 16384u) * sizeof(float);  // 193 KB
  k_attn<<<ag, 256, lds_bytes, stream>>>(
      q_abs, q_rope_, kv_nope, k_rope, kv_new, m_part, s_part, o_part);
  k_reduce<<<512, 128, 0, stream>>>(m_part, s_part, o_part, attn);
  k_out<<<32, 128, 0, stream>>>(attn, fused_ov, out);
}